// Net_35407710388877
// MI455X (gfx1250) — compile-verified
//
#include <hip/hip_runtime.h>
#include <hip/hip_bf16.h>
#include <math.h>

typedef __attribute__((ext_vector_type(2))) float v2f;
typedef __attribute__((ext_vector_type(8))) float v8f;

#define HMM_N 32
#define HMM_T 4096
#define HMM_K 128
#define HMM_V 50000

// ---------------------------------------------------------------------------
// K1: Texp[k,k'] = softmax_{k'}(w[k*K + k'])   (== exp(log_softmax))
// ---------------------------------------------------------------------------
__global__ __launch_bounds__(128) void trans_softmax(const float* __restrict__ w,
                                                     float* __restrict__ Texp) {
    const int k = blockIdx.x;
    const int j = threadIdx.x;
    __shared__ float red[128];
    float v = w[k * HMM_K + j];
    red[j] = v;
    __syncthreads();
    for (int s = 64; s > 0; s >>= 1) {
        if (j < s) red[j] = fmaxf(red[j], red[j + s]);
        __syncthreads();
    }
    float m = red[0];
    __syncthreads();
    float e = __expf(v - m);
    red[j] = e;
    __syncthreads();
    for (int s = 64; s > 0; s >>= 1) {
        if (j < s) red[j] += red[j + s];
        __syncthreads();
    }
    Texp[k * HMM_K + j] = e / red[0];
}

// ---------------------------------------------------------------------------
// K2: emission logits  E[v,k] = dot(emb[k,:], vocab[v,:])  via f32 WMMA.
// One wave per 16x16 tile. Output stored transposed (V,K).
// ---------------------------------------------------------------------------
__global__ __launch_bounds__(32) void emis_gemm(const float* __restrict__ emb,   // (K,K)
                                                const float* __restrict__ vocab, // (V,K)
                                                float* __restrict__ E) {         // (V,K)
    const int vi = blockIdx.x;          // vocab tile: 0..3124
    const int ki = blockIdx.y;          // cluster tile: 0..7
    const int lane   = threadIdx.x;
    const int lane15 = lane & 15;
    const int hi     = lane >> 4;       // which half-wave

    const int am = ki * 16 + lane15;    // cluster row for A fragment
    const int bn = vi * 16 + lane15;    // vocab column for B fragment

    v8f acc = {};
    #pragma unroll 4
    for (int kk = 0; kk < HMM_K; kk += 4) {
        const int j0 = kk + hi * 2;     // inner-dim index per ISA A/B layout
        v2f a, b;
        a.x = emb[am * HMM_K + j0];
        a.y = emb[am * HMM_K + j0 + 1];
        b.x = vocab[bn * HMM_K + j0];   // B[j, v] = vocab[v*K + j]
        b.y = vocab[bn * HMM_K + j0 + 1];
        acc = __builtin_amdgcn_wmma_f32_16x16x4_f32(false, a, false, b,
                                                    (short)0, acc, false, false);
    }
    // D layout: reg r -> tile row (hi*8 + r), col lane15. Store transposed.
    #pragma unroll
    for (int r = 0; r < 8; ++r) {
        const int k = ki * 16 + hi * 8 + r;
        const int v = vi * 16 + lane15;
        E[(size_t)v * HMM_K + k] = acc[r];
    }
}

// ---------------------------------------------------------------------------
// K3: lse[k] = logsumexp over V of E[:,k]
// ---------------------------------------------------------------------------
__global__ __launch_bounds__(256) void vocab_lse(const float* __restrict__ E,
                                                 float* __restrict__ lse) {
    const int k = blockIdx.x;
    const int tid = threadIdx.x;
    __shared__ float red[256];
    float m = -INFINITY;
    for (int v = tid; v < HMM_V; v += 256) m = fmaxf(m, E[(size_t)v * HMM_K + k]);
    red[tid] = m;
    __syncthreads();
    for (int s = 128; s > 0; s >>= 1) {
        if (tid < s) red[tid] = fmaxf(red[tid], red[tid + s]);
        __syncthreads();
    }
    m = red[0];
    __syncthreads();
    float sum = 0.f;
    for (int v = tid; v < HMM_V; v += 256) sum += __expf(E[(size_t)v * HMM_K + k] - m);
    red[tid] = sum;
    __syncthreads();
    for (int s = 128; s > 0; s >>= 1) {
        if (tid < s) red[tid] += red[tid + s];
        __syncthreads();
    }
    if (tid == 0) lse[k] = m + __logf(red[0]);
}

// ---------------------------------------------------------------------------
// K4: persistent single-workgroup forward scan.
// 512 threads = 16 waves; wave w owns output tile (mi = w>>3 over batch,
// ni = w&7 over k'). B fragments (exp(tran) column slice) live in registers
// for the whole scan. Per step: fused rowmax+exp (shfl butterflies, no LDS
// tree) -> 32 chained f32 WMMAs (A from LDS, B from regs) -> log + emission
// gather -> alpha writeback. Only 2 barriers per step.
// ---------------------------------------------------------------------------
__global__ __launch_bounds__(512) void hmm_scan(const int* __restrict__ x,        // (N,T)
                                                const float* __restrict__ start_w,
                                                const float* __restrict__ start_b,
                                                const float* __restrict__ Texp,   // (K,K)
                                                const float* __restrict__ lse,    // (K)
                                                const float* __restrict__ E,      // (V,K)
                                                float* __restrict__ out) {
    __shared__ float sAlpha[HMM_N * HMM_K];   // 16 KB
    __shared__ float sP[HMM_N * HMM_K];       // 16 KB
    __shared__ float sM[HMM_N];
    __shared__ int   sTok[HMM_N];
    __shared__ float sRow[HMM_N];

    const int tid = threadIdx.x;
    for (int i = tid; i < HMM_N * HMM_K; i += 512) {
        const int k = i & (HMM_K - 1);
        sAlpha[i] = start_w[k] + start_b[k];
    }

    const int wave   = tid >> 5;    // 0..15
    const int lane   = tid & 31;
    const int mi     = wave >> 3;   // batch tile 0..1
    const int ni     = wave & 7;    // k' tile 0..7
    const int lane15 = lane & 15;
    const int hi     = lane >> 4;
    const int am     = mi * 16 + lane15;   // A-fragment batch row
    const int bn     = ni * 16 + lane15;   // B-fragment / output column k'

    // Per-lane constants for the whole scan.
    const float lse_kp = lse[bn];

    // B fragments: exp(tran)[:, bn] column slice, register-resident (64 f32).
    v2f bfrag[32];
    #pragma unroll
    for (int i = 0; i < 32; ++i) {
        const int j0 = i * 4 + hi * 2;
        bfrag[i].x = Texp[j0 * HMM_K + bn];
        bfrag[i].y = Texp[(j0 + 1) * HMM_K + bn];
    }
    __syncthreads();

    // Fused-phase indices: 16 consecutive lanes (same wave half) own one row.
    const int rown = tid >> 4;     // 0..31 batch row this thread helps reduce
    const int rsub = tid & 15;     // slot within the 16-thread row group

    for (int t = 0; t < HMM_T; ++t) {
        // tokens for this step + prefetch next step's emission rows
        if (tid < HMM_N) {
            sTok[tid] = x[tid * HMM_T + t];
            if (t + 1 < HMM_T) {
                const int nxt = x[tid * HMM_T + t + 1];
                __builtin_prefetch(&E[(size_t)nxt * HMM_K], 0, 1);
            }
        }
        // fused rowmax (shfl butterfly over 16 lanes) + exp into sP
        {
            float a0[8];
            const float* row = &sAlpha[rown * HMM_K + rsub * 8];
            float m = row[0];
            a0[0] = row[0];
            #pragma unroll
            for (int j = 1; j < 8; ++j) { a0[j] = row[j]; m = fmaxf(m, a0[j]); }
            m = fmaxf(m, __shfl_xor(m, 1));
            m = fmaxf(m, __shfl_xor(m, 2));
            m = fmaxf(m, __shfl_xor(m, 4));
            m = fmaxf(m, __shfl_xor(m, 8));
            if (rsub == 0) sM[rown] = m;
            float* prow = &sP[rown * HMM_K + rsub * 8];
            #pragma unroll
            for (int j = 0; j < 8; ++j) prow[j] = __expf(a0[j] - m);
        }
        __syncthreads();   // B1: sP, sTok, sM ready

        // emission gathers issued early to overlap the WMMA chain
        float eg[8];
        #pragma unroll
        for (int r = 0; r < 8; ++r) {
            const int n = mi * 16 + hi * 8 + r;
            eg[r] = E[(size_t)sTok[n] * HMM_K + bn];
        }

        // G(16x16) = P[tile mi] x Texp[:, tile ni]: 32 chained f32 WMMAs
        v8f acc = {};
        #pragma unroll
        for (int i = 0; i < 32; ++i) {
            const int j0 = i * 4 + hi * 2;
            v2f a;
            a.x = sP[am * HMM_K + j0];
            a.y = sP[am * HMM_K + j0 + 1];
            acc = __builtin_amdgcn_wmma_f32_16x16x4_f32(false, a, false, bfrag[i],
                                                        (short)0, acc, false, false);
        }

        // alpha' = m[n] + log(G) + E[tok[n], k'] - lse[k']
        #pragma unroll
        for (int r = 0; r < 8; ++r) {
            const int n = mi * 16 + hi * 8 + r;
            sAlpha[n * HMM_K + bn] = sM[n] + __logf(acc[r]) + eg[r] - lse_kp;
        }
        __syncthreads();   // B2: alpha complete for next step
    }

    // final: -mean_n logsumexp_k alpha_T
    if (tid < HMM_N) {
        const float* row = &sAlpha[tid * HMM_K];
        float m = row[0];
        for (int j = 1; j < HMM_K; ++j) m = fmaxf(m, row[j]);
        float s = 0.f;
        for (int j = 0; j < HMM_K; ++j) s += __expf(row[j] - m);
        sRow[tid] = m + __logf(s);
    }
    __syncthreads();
    if (tid == 0) {
        float s = 0.f;
        for (int n = 0; n < HMM_N; ++n) s += sRow[n];
        out[0] = -s / (float)HMM_N;
    }
}

// ---------------------------------------------------------------------------
extern "C" void kernel_launch(void* const* d_in, const int* in_sizes, int n_in,
                              void* d_out, int out_size, void* d_ws, size_t ws_size,
                              hipStream_t stream) {
    const int*   x        = (const int*)d_in[0];     // (N,T) tokens
    const float* start_w  = (const float*)d_in[1];   // (K,1)
    const float* start_b  = (const float*)d_in[2];   // (K)
    const float* trans_w  = (const float*)d_in[3];   // (K*K,1)
    const float* emb_w    = (const float*)d_in[4];   // (K,K)
    const float* vocab_w  = (const float*)d_in[5];   // (V,K)

    float* ws   = (float*)d_ws;
    float* Texp = ws;                                 // K*K floats @ 0
    float* lseb = ws + HMM_K * HMM_K;                 // K floats @ 64KB
    float* E    = ws + 32768;                         // V*K floats @ 128KB

    trans_softmax<<<HMM_K, 128, 0, stream>>>(trans_w, Texp);

    dim3 g2(HMM_V / 16, HMM_K / 16);
    emis_gemm<<<g2, 32, 0, stream>>>(emb_w, vocab_w, E);

    vocab_lse<<<HMM_K, 256, 0, stream>>>(E, lseb);

    hmm_scan<<<1, 512, 0, stream>>>(x, start_w, start_b, Texp, lseb, E,
                                    (float*)d_out);
}